// HardBinaryVote_36515811950592
// MI455X (gfx1250) — compile-verified
//
#include <hip/hip_runtime.h>
#include <hip/hip_bf16.h>

// HardBinaryVote on MI455X (gfx1250):
// out[c] = (sum_v in[v][c] >= 16) for 31 voters -> pure HBM-bandwidth problem
// (~256 MB @ 23.3 TB/s ~= 11 us). Uses the CDNA5 async global->LDS copy path
// (GLOBAL_LOAD_ASYNC_TO_LDS_B128, ASYNCcnt split waits, NT temporal hint) as a
// VGPR-free deep load pipeline, double-buffered in LDS.

#define NV   31     // voters
#define CH   8      // voters per pipeline chunk (chunks: 8,8,8,7)
#define TPB  256    // threads per block (8 wave32)

__device__ __forceinline__ void async_b128(unsigned lds_off, unsigned voff,
                                           const int* base) {
  // GVS mode: mem_addr = SGPR_U64(base) + VGPR_I32(voff); dest LDS byte addr in VGPR.
  asm volatile("global_load_async_to_lds_b128 %0, %1, %2 th:TH_LOAD_NT"
               :: "v"(lds_off), "v"(voff), "s"(base)
               : "memory");
}

template <int C, int BUF, int N>
__device__ __forceinline__ void issue_chunk(const int* in, unsigned cb,
                                            unsigned vs, unsigned lds_base) {
#pragma unroll
  for (int k = 0; k < N; ++k) {
    unsigned voff = cb + (unsigned)(C * CH + k) * vs;                 // global byte offset
    unsigned loff = lds_base + (unsigned)(BUF * CH + k) * (TPB * 16); // LDS byte offset
    async_b128(loff, voff, in);
  }
}

template <int BUF, int N>
__device__ __forceinline__ void read_chunk(const int4* sbuf, int tid, int4& acc) {
#pragma unroll
  for (int k = 0; k < N; ++k) {
    int4 v = sbuf[(BUF * CH + k) * TPB + tid]; // ds_load_b128, conflict-free
    acc.x += v.x; acc.y += v.y; acc.z += v.z; acc.w += v.w;
  }
}

__global__ __launch_bounds__(TPB) void HardBinaryVote_kernel(
    const int* __restrict__ in, float* __restrict__ out, int B) {
  __shared__ int4 sbuf[2 * CH * TPB]; // 64 KiB -> 5 blocks / WGP vs 320 KiB LDS

  const int tid = (int)threadIdx.x;
  unsigned col = ((unsigned)blockIdx.x * TPB + (unsigned)tid) * 4u;
  const unsigned maxcol = (unsigned)(B - 4);
  if (col > maxcol) col = maxcol; // clamp tail: duplicate deterministic work, EXEC stays full

  const unsigned cb = col * 4u;         // column byte offset  (< 8e6, fits i32)
  const unsigned vs = (unsigned)B * 4u; // voter stride, bytes (8e6; 31*vs < 2^31)

  // Flat address of an LDS object has the LDS byte offset in its low 32 bits.
  const unsigned lds_base =
      (unsigned)(unsigned long long)(&sbuf[0]) + (unsigned)tid * 16u;

  int4 acc = {0, 0, 0, 0};

  // ---- software pipeline: 4 chunks, 2 LDS buffers, <=16 async B128 in flight ----
  issue_chunk<0, 0, CH>(in, cb, vs, lds_base);          // outstanding: 8
  issue_chunk<1, 1, CH>(in, cb, vs, lds_base);          // outstanding: 16

  asm volatile("s_wait_asynccnt 8" ::: "memory");       // chunk0 landed (in-order)
  read_chunk<0, CH>(sbuf, tid, acc);                    // voters 0..7
  asm volatile("s_wait_dscnt 0" ::: "memory");          // buf0 reads done before reuse
  issue_chunk<2, 0, CH>(in, cb, vs, lds_base);          // outstanding: <=16

  asm volatile("s_wait_asynccnt 8" ::: "memory");       // chunk1 landed
  read_chunk<1, CH>(sbuf, tid, acc);                    // voters 8..15
  asm volatile("s_wait_dscnt 0" ::: "memory");          // buf1 reads done before reuse
  issue_chunk<3, 1, NV - 3 * CH>(in, cb, vs, lds_base); // 7 loads, outstanding: <=15

  asm volatile("s_wait_asynccnt 7" ::: "memory");       // chunk2 landed
  read_chunk<0, CH>(sbuf, tid, acc);                    // voters 16..23

  asm volatile("s_wait_asynccnt 0" ::: "memory");       // chunk3 landed
  read_chunk<1, NV - 3 * CH>(sbuf, tid, acc);           // voters 24..30

  // 31 voters: majority <=> count_of_ones >= 16 (tie impossible; matches argmax)
  float4 r;
  r.x = (acc.x >= 16) ? 1.0f : 0.0f;
  r.y = (acc.y >= 16) ? 1.0f : 0.0f;
  r.z = (acc.z >= 16) ? 1.0f : 0.0f;
  r.w = (acc.w >= 16) ? 1.0f : 0.0f;
  *(float4*)(out + col) = r;
}

extern "C" void kernel_launch(void* const* d_in, const int* in_sizes, int n_in,
                              void* d_out, int out_size, void* d_ws, size_t ws_size,
                              hipStream_t stream) {
  const int* in = (const int*)d_in[0];   // [31, B] int32 votes in {0,1}
  float* out = (float*)d_out;            // B elements
  const int B = out_size;                // 2,000,000 (== in_sizes[0] / 31)
  const int threads = (B + 3) / 4;       // one int4 (4 columns) per thread
  const int blocks = (threads + TPB - 1) / TPB;
  hipLaunchKernelGGL(HardBinaryVote_kernel, dim3(blocks), dim3(TPB), 0, stream,
                     in, out, B);
}